// Patcher3D_11166914969753
// MI455X (gfx1250) — compile-verified
//
#include <hip/hip_runtime.h>

// CDNA5 (gfx1250) fused 2-level 3D Haar DWT via V_WMMA_F32_16X16X4_F32.
// in : x[4,3,33,256,256] f32   out: [4,192,9,64,64] f32
// Each wave computes one (b,c,t,h,wtile16): OUT[64x16] = M[64x64] * X[64x16]
// with M the +/-1 (H(x)H(x)H) sign matrix, scale 1/64 applied at store.

typedef __attribute__((ext_vector_type(2))) float v2f;
typedef __attribute__((ext_vector_type(8))) float v8f;

#define NB   4      // batch
#define NC   3      // channels in
#define TIN  33
#define HIN  256
#define WIN  256
#define TOUT 9
#define HOUT 64
#define WOUT 64
#define CH_STRIDE (TOUT*HOUT*WOUT)   // 36864 elements per output channel

__global__ __launch_bounds__(256) void haar3d_p4_wmma(
    const float* __restrict__ x, float* __restrict__ out)
{
  const int lane = threadIdx.x & 31;
  const int wave = threadIdx.x >> 5;
  int task = blockIdx.x * 8 + wave;          // 27648 wave-tasks total, exact

  const int wt = task & 3;   task >>= 2;     // w tile (16 outputs)
  const int h  = task & 63;  task >>= 6;     // output h
  const int t  = task % 9;   task /= 9;      // output t
  const int c  = task % 3;
  const int b  = task / 3;

  const int lhalf = lane >> 4;               // 0: lanes 0-15, 1: lanes 16-31
  const int lmod  = lane & 15;
  // B(4x16) ISA layout: VGPR0 holds K=0 (lanes 0-15) / K=2 (lanes 16-31),
  // VGPR1 holds K=1 / K=3.  A per-lane b64 load at this element offset of a
  // contiguous 64-float row yields that layout directly (256B coalesced).
  const int laneOff = lmod * 4 + lhalf * 2;

  // ---- preload all 16 B operand slices (one per (it,ih) input row) ----
  v2f bv[16];
#pragma unroll
  for (int k = 0; k < 16; ++k) {
    const int it  = k >> 2;
    const int ih  = k & 3;
    const int te  = 4 * t + it;              // extended temporal index [0,36)
    const int tin = (te < 4) ? 0 : (te - 3); // frame-0 repeated 4x
    const int hin = 4 * h + ih;
    const int rowBase = (((b * NC + c) * TIN + tin) * HIN + hin) * WIN + wt * 64;
    bv[k] = *reinterpret_cast<const v2f*>(x + rowBase + laneOff);
  }

  v8f acc[4];
#pragma unroll
  for (int q = 0; q < 4; ++q) acc[q] = {};

  // ---- 64 WMMAs: OUT[64x16] = M * X, M regenerated as +/-1 per slice ----
  // input offset i = 16*it + 4*ih + iw has bits [k3 k2 k1 k0 | hi b0];
  // parity(o,i) = popc(o & g(i)) with g = k3<<5|k1<<4|hi<<3|k2<<2|k0<<1|b0.
#pragma unroll
  for (int k = 0; k < 16; ++k) {
    const int g0 = (((k >> 3) & 1) << 5) | (((k >> 1) & 1) << 4) | (lhalf << 3)
                 | (((k >> 2) & 1) << 2) | ((k & 1) << 1);
#pragma unroll
    for (int q = 0; q < 4; ++q) {
      const int o = q * 16 + lmod;           // A row = output index o
      const unsigned s0 = (unsigned)(__popc(o & g0) & 1);
      const unsigned s1 = s0 ^ (unsigned)(o & 1);   // g1 = g0 | 1
      v2f av;                                        // A 16x4: VGPR0=K(2*lhalf), VGPR1=+1
      av.x = __uint_as_float(0x3F800000u | (s0 << 31));
      av.y = __uint_as_float(0x3F800000u | (s1 << 31));
      acc[q] = __builtin_amdgcn_wmma_f32_16x16x4_f32(
          false, av, false, bv[k], (short)0, acc[q], false, false);
    }
  }

  // ---- scatter to 64 output channels; per-VGPR store is 2x64B coalesced ----
  const float scale = 1.0f / 64.0f;
  const int base0 = (b * 192 + c) * CH_STRIDE + t * (HOUT * WOUT)
                  + h * WOUT + wt * 16 + lmod;
#pragma unroll
  for (int q = 0; q < 4; ++q) {
#pragma unroll
    for (int v = 0; v < 8; ++v) {
      const int o = q * 16 + v + 8 * lhalf;  // D: VGPR v -> M=v (lo) / M=8+v (hi)
      const int chOff = ((o >> 3) * 24 + (o & 7) * 3) * CH_STRIDE;
      out[base0 + chOff] = acc[q][v] * scale;
    }
  }
}

extern "C" void kernel_launch(void* const* d_in, const int* in_sizes, int n_in,
                              void* d_out, int out_size, void* d_ws, size_t ws_size,
                              hipStream_t stream) {
  (void)in_sizes; (void)n_in; (void)d_ws; (void)ws_size; (void)out_size;
  const float* x = (const float*)d_in[0];   // d_in[1] = patch_size (==4, fused)
  float* out = (float*)d_out;
  const int waveTasks = NB * NC * TOUT * HOUT * (WOUT / 16);  // 27648
  dim3 grid(waveTasks / 8), block(256);
  hipLaunchKernelGGL(haar3d_p4_wmma, grid, block, 0, stream, x, out);
}